// DetectionLoss_49692771615067
// MI455X (gfx1250) — compile-verified
//
#include <hip/hip_runtime.h>
#include <hip/hip_bf16.h>
#include <stdint.h>

// Problem dims (fixed by reference setup_inputs)
#define B_  16
#define C_  80
#define H_  160
#define W_  160
#define N_  32
#define CHW (C_*H_*W_)            // 2,048,000 elements per batch slab

// TDM staging config: 4096 f32 per chunk (16 KB), double buffered (32 KB LDS)
#define CHUNK 4096
#define NCHUNK 4
#define BLOCKS_PER_BATCH 125      // 125 * 4 * 4096 == 2,048,000
#define THREADS 256               // 8 wave32 per block

typedef unsigned int u32x4 __attribute__((ext_vector_type(4)));
typedef int          i32x8 __attribute__((ext_vector_type(8)));
typedef int          i32x4 __attribute__((ext_vector_type(4)));

// Workspace layout (floats):
// [0..15]  per-batch total softplus sum (atomicAdd target, must be zeroed)
// [16..31] per-batch pos_sum   = sum_unique(softplus(xp) - xp)
// [32..47] per-batch pos_cnt   (as float)
// [48..63] per-batch pos_sp    = sum_unique(softplus(xp))
// [64..79] per-batch bbox L1 sum

__device__ __forceinline__ float softplus_fast(float x) {
  // logaddexp(0,x) = max(x,0) + log(1 + exp(-|x|)); hw v_exp_f32 / v_log_f32
  return fmaxf(x, 0.f) + __logf(1.f + __expf(-fabsf(x)));
}

// Issue one TDM descriptor: copy CHUNK f32 (1 row tile) from global -> LDS.
// D# bit packing per CDNA5 ISA ch.8 (group0: count/lds_addr/global_addr/type,
// group1: data_size=4B, tensor_dim0=tile_dim0=CHUNK, tile_dim1=1).
// This toolchain (clang-23 / therock-10.0) exposes the 6-arg builtin form.
__device__ __forceinline__ void tdm_load_chunk(uint32_t lds_addr, const float* gptr) {
  uint64_t ga = (uint64_t)(uintptr_t)gptr;
  u32x4 g0;
  g0.x = 1u;                                   // count=1, user descriptor
  g0.y = lds_addr;                             // LDS byte address
  g0.z = (uint32_t)(ga & 0xFFFFFFFFu);         // global_addr[31:0]
  g0.w = (uint32_t)(ga >> 32) | (2u << 30);    // global_addr[56:32] | type=2
  i32x8 g1;
  g1[0] = (int)(2u << 16);                     // workgroup_mask=0, data_size=2 (4B)
  g1[1] = (int)(((uint32_t)CHUNK & 0xFFFFu) << 16);          // tensor_dim0 lo16
  g1[2] = (int)(((uint32_t)CHUNK >> 16) | (1u << 16));       // tensor_dim0 hi | tensor_dim1=1 lo
  g1[3] = (int)((uint32_t)CHUNK << 16);        // tensor_dim1 hi=0 | tile_dim0=CHUNK
  g1[4] = 1;                                   // tile_dim1=1, tile_dim2=0
  g1[5] = CHUNK;                               // tensor_dim0_stride lo32
  g1[6] = 0;                                   // stride0 hi16 | stride1 lo16
  g1[7] = 0;                                   // stride1 hi32
  i32x4 z4 = {0, 0, 0, 0};                     // groups 2/3 unused (<=2D tensor)
  i32x8 z8 = {0, 0, 0, 0, 0, 0, 0, 0};         // extra group (6-arg toolchain form)
  __builtin_amdgcn_tensor_load_to_lds(g0, g1, z4, z4, z8, 0);
}

// ---------------------------------------------------------------------------
// Kernel 0: zero the atomic accumulation slots (harness poisons d_ws)
__global__ void dl_init_kernel(float* ws) {
  if (threadIdx.x < 16) ws[threadIdx.x] = 0.f;
}

// ---------------------------------------------------------------------------
// Kernel 1: per-batch sum of softplus(cls_pred) with TDM-staged double buffer
__global__ __launch_bounds__(THREADS)
void dl_softplus_reduce_kernel(const float* __restrict__ cls_pred,
                               float* __restrict__ ws) {
  __shared__ float smem[2 * CHUNK];
  __shared__ float wsum[THREADS / 32];

  const int b    = blockIdx.y;
  const int blk  = blockIdx.x;
  const int tid  = threadIdx.x;
  const int wave = tid >> 5;

  const float* base = cls_pred + (size_t)b * CHW + (size_t)blk * (NCHUNK * CHUNK);
  const uint32_t smem_base = (uint32_t)(uintptr_t)&smem[0]; // low 32 bits = LDS offset

  if (wave == 0) tdm_load_chunk(smem_base, base);           // prologue: chunk 0

  // 4 independent accumulators: keep the v_exp/v_log pipeline full instead of
  // serializing dependent v_add_f32 behind transcendental latency.
  float a0 = 0.f, a1 = 0.f, a2 = 0.f, a3 = 0.f;
  for (int i = 0; i < NCHUNK; ++i) {
    if (wave == 0) {
      if (i + 1 < NCHUNK) {
        tdm_load_chunk(smem_base + (uint32_t)(((i + 1) & 1) * (CHUNK * 4)),
                       base + (size_t)(i + 1) * CHUNK);
        __builtin_amdgcn_s_wait_tensorcnt(1);               // chunk i landed
      } else {
        __builtin_amdgcn_s_wait_tensorcnt(0);               // last chunk landed
      }
    }
    __syncthreads();                                        // publish LDS to all waves
    const float4* lv = (const float4*)(smem + (i & 1) * CHUNK);
#pragma unroll
    for (int k = 0; k < CHUNK / (4 * THREADS); ++k) {       // 4 x ds_load_b128 / thread
      float4 v = lv[k * THREADS + tid];
      a0 += softplus_fast(v.x);
      a1 += softplus_fast(v.y);
      a2 += softplus_fast(v.z);
      a3 += softplus_fast(v.w);
    }
    __syncthreads();                                        // buffer safe to overwrite
  }
  float acc = (a0 + a1) + (a2 + a3);

  // wave32 shuffle reduction, then cross-wave via LDS
  for (int off = 16; off; off >>= 1) acc += __shfl_xor(acc, off, 32);
  if ((tid & 31) == 0) wsum[wave] = acc;
  __syncthreads();
  if (tid == 0) {
    float s = 0.f;
#pragma unroll
    for (int w = 0; w < THREADS / 32; ++w) s += wsum[w];
    atomicAdd(&ws[b], s);
  }
}

// ---------------------------------------------------------------------------
// Kernel 2: per-(b,n) assignment, duplicate dedup within one wave32, gathers
__global__ __launch_bounds__(32)
void dl_assign_kernel(const float* __restrict__ cls_pred,
                      const float* __restrict__ bbox_pred,
                      const float* __restrict__ gt_bboxes,
                      const int*   __restrict__ gt_labels,
                      float* __restrict__ ws) {
  const int b = blockIdx.x;    // 16 blocks
  const int n = threadIdx.x;   // 32 lanes == N_

  float4 gb = ((const float4*)gt_bboxes)[b * N_ + n];
  float x1 = gb.x * (float)W_, y1 = gb.y * (float)H_;
  float x2 = gb.z * (float)W_, y2 = gb.w * (float)H_;
  float cx = fminf(fmaxf((x1 + x2) * 0.5f, 0.f), (float)(W_ - 1));
  float cy = fminf(fmaxf((y1 + y2) * 0.5f, 0.f), (float)(H_ - 1));
  int gx = (int)cx, gy = (int)cy;           // toward-zero, non-negative
  int lbl = gt_labels[b * N_ + n];
  int key = (lbl * H_ + gy) * W_ + gx;      // flat (c,h,w) index, < 2,048,000

  // ".at[].set(1.0)": positives are a SET -> keep only first occurrence
  bool dup = false;
#pragma unroll
  for (int j = 0; j < N_; ++j) {
    int kj = __shfl(key, j, 32);
    if (j < n && kj == key) dup = true;
  }

  float xp = cls_pred[(size_t)b * CHW + (size_t)key];
  float sp = fmaxf(xp, 0.f) + log1pf(expf(-fabsf(xp)));
  float u       = dup ? 0.f : 1.f;          // unique-positive indicator
  float pos_sum = u * (sp - xp);            // bce at target==1
  float pos_sp  = u * sp;                   // softplus to subtract from neg side

  // bbox L1: pred_boxes = bbox_pred[b, :, gy, gx] (all n, duplicates included)
  const float* bp = bbox_pred + ((size_t)b * 4 * H_ + gy) * W_ + gx;
  float bb = 0.25f * (fabsf(bp[0 * H_ * W_] - x1) +
                      fabsf(bp[1 * H_ * W_] - y1) +
                      fabsf(bp[2 * H_ * W_] - x2) +
                      fabsf(bp[3 * H_ * W_] - y2));

  for (int off = 16; off; off >>= 1) {
    u       += __shfl_xor(u,       off, 32);
    pos_sum += __shfl_xor(pos_sum, off, 32);
    pos_sp  += __shfl_xor(pos_sp,  off, 32);
    bb      += __shfl_xor(bb,      off, 32);
  }
  if (n == 0) {
    ws[16 + b] = pos_sum;
    ws[32 + b] = u;
    ws[48 + b] = pos_sp;
    ws[64 + b] = bb;
  }
}

// ---------------------------------------------------------------------------
// Kernel 3: final scalar combine
__global__ __launch_bounds__(32)
void dl_combine_kernel(const float* __restrict__ ws, float* __restrict__ out) {
  const int t = threadIdx.x;
  float pos_term = 0.f, neg_term = 0.f, bb = 0.f;
  if (t < B_) {
    float total_sp = ws[t];
    float pos_sum  = ws[16 + t];
    float pos_cnt  = ws[32 + t];
    float pos_sp   = ws[48 + t];
    pos_term = pos_sum / pos_cnt;
    neg_term = (total_sp - pos_sp) / ((float)CHW - pos_cnt);
    bb       = ws[64 + t];
  }
  for (int off = 16; off; off >>= 1) {
    pos_term += __shfl_xor(pos_term, off, 32);
    neg_term += __shfl_xor(neg_term, off, 32);
    bb       += __shfl_xor(bb,       off, 32);
  }
  if (t == 0) {
    const float num_pos  = (float)(B_ * N_);
    float pos_cls  = pos_term / num_pos;
    float neg_cls  = neg_term / (float)B_;
    float bbox_l   = bb / num_pos;
    float cls_loss = pos_cls + 0.25f * neg_cls;
    float total    = 0.5f * cls_loss + 7.5f * bbox_l + 1e-6f;
    out[0] = total;
    out[1] = cls_loss;
    out[2] = bbox_l;
  }
}

// ---------------------------------------------------------------------------
extern "C" void kernel_launch(void* const* d_in, const int* in_sizes, int n_in,
                              void* d_out, int out_size, void* d_ws, size_t ws_size,
                              hipStream_t stream) {
  const float* cls_pred  = (const float*)d_in[0];
  const float* bbox_pred = (const float*)d_in[1];
  const float* gt_bboxes = (const float*)d_in[2];
  const int*   gt_labels = (const int*)  d_in[3];
  float* out = (float*)d_out;
  float* ws  = (float*)d_ws;

  dl_init_kernel<<<1, 32, 0, stream>>>(ws);
  dl_softplus_reduce_kernel<<<dim3(BLOCKS_PER_BATCH, B_), THREADS, 0, stream>>>(cls_pred, ws);
  dl_assign_kernel<<<B_, 32, 0, stream>>>(cls_pred, bbox_pred, gt_bboxes, gt_labels, ws);
  dl_combine_kernel<<<1, 32, 0, stream>>>(ws, out);
}